// FutureEncoder_54778012893467
// MI455X (gfx1250) — compile-verified
//
#include <hip/hip_runtime.h>

typedef float v2f __attribute__((ext_vector_type(2)));
typedef float v8f __attribute__((ext_vector_type(8)));

#define FUTURE_K 32
#define NEG_INF  (-1000000000.0f)

// One wave (32 lanes) per workgroup handles one tile of 16 consecutive query rows.
// scores: three 16x16 f32 WMMA accumulators over key tiles at offsets +1, +17, +33.
// softmax: band-masked, half-wave shuffle reductions (wave32: lanes 0-15 / 16-31).
// output: W(16x48) x FW(48xD) via f32 WMMA, K-chunks of 4, weights staged in LDS.
__launch_bounds__(32)
__global__ void future_encoder_kernel(const float* __restrict__ h,
                                      float* __restrict__ out,
                                      int B, int S, int D) {
    __shared__ float w_lds[16 * 48];

    const int tilesPerBatch = S >> 4;
    const int tile = blockIdx.x;
    const int b    = tile / tilesPerBatch;
    const int s0   = (tile % tilesPerBatch) << 4;

    const int lane = threadIdx.x;   // 0..31
    const int half = lane >> 4;     // 0 or 1 (selects K-pair / M-group)
    const int lm   = lane & 15;     // row/col within tile

    const float* __restrict__ hb = h + (size_t)b * S * D;

    // ---------------- Phase 1: scores = Q x K^T (three 16x16 tiles) -------------
    v8f c0 = {0.f,0.f,0.f,0.f,0.f,0.f,0.f,0.f};
    v8f c1 = c0, c2 = c0;

    // A layout (16x4 f32): lane holds row M = lm, k = 2*half + {0,1}
    const int aRow = s0 + lm;
    // B layout (4x16 f32): lane holds col N = lm (key), k = 2*half + {0,1}
    int kRow0 = s0 + 1 +  0 + lm; if (kRow0 > S - 1) kRow0 = S - 1;
    int kRow1 = s0 + 1 + 16 + lm; if (kRow1 > S - 1) kRow1 = S - 1;
    int kRow2 = s0 + 1 + 32 + lm; if (kRow2 > S - 1) kRow2 = S - 1;

    const float* pA  = hb + (size_t)aRow  * D + 2 * half;
    const float* pB0 = hb + (size_t)kRow0 * D + 2 * half;
    const float* pB1 = hb + (size_t)kRow1 * D + 2 * half;
    const float* pB2 = hb + (size_t)kRow2 * D + 2 * half;

    for (int d0 = 0; d0 < D; d0 += 4) {
        v2f a  = *(const v2f*)(pA  + d0);
        v2f b0 = *(const v2f*)(pB0 + d0);
        v2f b1 = *(const v2f*)(pB1 + d0);
        v2f b2 = *(const v2f*)(pB2 + d0);
        c0 = __builtin_amdgcn_wmma_f32_16x16x4_f32(false, a, false, b0, (short)0, c0, false, false);
        c1 = __builtin_amdgcn_wmma_f32_16x16x4_f32(false, a, false, b1, (short)0, c1, false, false);
        c2 = __builtin_amdgcn_wmma_f32_16x16x4_f32(false, a, false, b2, (short)0, c2, false, false);
    }

    // ---------------- Phase 2: band mask + softmax over 48 candidate keys -------
    // C/D layout: lane holds col N = lm; VGPR v holds row q = v + 8*half.
    float w0[8], w1[8], w2[8];
    #pragma unroll
    for (int v = 0; v < 8; ++v) {
        const int q = v + 8 * half;
        float s[3] = { c0[v], c1[v], c2[v] };
        bool ok[3];
        float m = NEG_INF;
        #pragma unroll
        for (int j = 0; j < 3; ++j) {
            const int delta  = 1 + 16 * j + lm - q;   // key - query distance
            const int keyrow = s0 + 1 + 16 * j + lm;
            ok[j] = (delta >= 1) && (delta <= FUTURE_K) && (keyrow < S);
            if (!ok[j]) s[j] = NEG_INF;
            m = fmaxf(m, s[j]);
        }
        // max across the 16 lanes of this half (xor masks < 16 stay in-half)
        #pragma unroll
        for (int off = 1; off < 16; off <<= 1)
            m = fmaxf(m, __shfl_xor(m, off, 32));
        float e[3];
        float sum = 0.f;
        #pragma unroll
        for (int j = 0; j < 3; ++j) {
            e[j] = ok[j] ? __expf(s[j] - m) : 0.f;
            sum += e[j];
        }
        #pragma unroll
        for (int off = 1; off < 16; off <<= 1)
            sum += __shfl_xor(sum, off, 32);
        const float inv = (sum > 0.f) ? (1.f / sum) : 0.f;  // all-invalid row -> 0
        w0[v] = e[0] * inv;
        w1[v] = e[1] * inv;
        w2[v] = e[2] * inv;
    }

    // Stage weights in LDS to re-read them in A-matrix (row-major-per-lane) layout.
    #pragma unroll
    for (int v = 0; v < 8; ++v) {
        const int q = v + 8 * half;
        w_lds[q * 48 +  0 + lm] = w0[v];
        w_lds[q * 48 + 16 + lm] = w1[v];
        w_lds[q * 48 + 32 + lm] = w2[v];
    }
    __syncthreads();

    // ---------------- Phase 3: out(16xD) = W(16x48) x FW(48xD) ------------------
    // A chunk layout: lane holds row lm, k = kk + 2*half + {0,1}
    const float* wrow = &w_lds[lm * 48 + 2 * half];
    float* __restrict__ ob = out + (size_t)b * S * D;

    for (int dt = 0; dt < D; dt += 16) {
        v8f acc = {0.f,0.f,0.f,0.f,0.f,0.f,0.f,0.f};
        #pragma unroll
        for (int kk = 0; kk < 48; kk += 4) {
            v2f a;
            a.x = wrow[kk];         // weight for key index kk + 2*half
            a.y = wrow[kk + 1];     // weight for key index kk + 2*half + 1
            int kr0 = s0 + 1 + kk + 2 * half;
            int kr1 = kr0 + 1;
            if (kr0 > S - 1) kr0 = S - 1;   // OOB keys carry zero weight
            if (kr1 > S - 1) kr1 = S - 1;
            v2f bb;
            bb.x = hb[(size_t)kr0 * D + dt + lm];
            bb.y = hb[(size_t)kr1 * D + dt + lm];
            acc = __builtin_amdgcn_wmma_f32_16x16x4_f32(false, a, false, bb, (short)0, acc, false, false);
        }
        #pragma unroll
        for (int v = 0; v < 8; ++v)
            ob[(size_t)(s0 + v + 8 * half) * D + dt + lm] = acc[v];
    }
}

extern "C" void kernel_launch(void* const* d_in, const int* in_sizes, int n_in,
                              void* d_out, int out_size, void* d_ws, size_t ws_size,
                              hipStream_t stream) {
    const float* h = (const float*)d_in[0];
    float* out = (float*)d_out;
    const int S = 2048, D = 1024;
    const int B = in_sizes[0] / (S * D);     // 4 for the reference shapes
    dim3 grid(B * (S / 16));
    dim3 block(32);
    hipLaunchKernelGGL(future_encoder_kernel, grid, block, 0, stream, h, out, B, S, D);
}